// diffusion_gcn_distance_graph_90872918049151
// MI455X (gfx1250) — compile-verified
//
#include <hip/hip_runtime.h>

#define N_NODES 207
#define NPAD    208        // padded N for 16x16 tiles / K padding (zeros)
#define LSEQ    2048
#define E_EDGES 1722
#define C_CH    2
#define B_BATCH 64
#define LDS_A_STRIDE 212   // 16 rows x 208 K, padded stride -> conflict-free b64 reads
#define LDS_X_STRIDE 18    // 256 cols x 16 K transposed, padded stride -> conflict-free b64 reads

typedef __attribute__((ext_vector_type(2))) float v2f;
typedef __attribute__((ext_vector_type(8))) float v8f;

// ---------------------------------------------------------------------------
// Kernel 1: zero the padded Laplacian workspace (B, NPAD, NPAD)
// ---------------------------------------------------------------------------
__global__ __launch_bounds__(256) void zero_lap_kernel(float* __restrict__ lap, int n) {
  int i = blockIdx.x * 256 + threadIdx.x;
  if (i < n) lap[i] = 0.0f;
}

// ---------------------------------------------------------------------------
// Kernel 2: scatter edges -> Lap = diag(colsum(W)) - W
//   for each edge e: Lap[r,c] -= v ; Lap[c,c] += v
// ---------------------------------------------------------------------------
__global__ __launch_bounds__(256) void scatter_lap_kernel(
    const int*   __restrict__ ind,
    const int*   __restrict__ rows,
    const int*   __restrict__ cols,
    const float* __restrict__ weight_diff,   // (T, E)
    float*       __restrict__ lap) {         // (B, NPAD, NPAD)
  const int b = blockIdx.x;
  float* lap_b = lap + (size_t)b * NPAD * NPAD;
  const int t = ind[b];                      // RES == 1 -> ii = ind
  const float* w = weight_diff + (size_t)t * E_EDGES;
  for (int e = threadIdx.x; e < E_EDGES; e += 256) {
    const float v = w[e];
    const int r = rows[e];
    const int c = cols[e];
    atomicAdd(lap_b + r * NPAD + c, -v);
    atomicAdd(lap_b + c * NPAD + c,  v);
  }
}

// ---------------------------------------------------------------------------
// Kernel 3: out[b] = Lap_b @ X_b + bias + X_b   via V_WMMA_F32_16X16X4_F32
//   grid: (L/256, NPAD/16, B), block: 256 threads = 8 waves
//   each wave computes two 16x16 output tiles
// ---------------------------------------------------------------------------
__global__ __launch_bounds__(256) void lap_gemm_kernel(
    const float* __restrict__ inputs,          // (B, C, N, L), use channel 0
    const int*   __restrict__ ind,
    const float* __restrict__ bias_diffusion,  // (T, N)
    const float* __restrict__ lap,             // (B, NPAD, NPAD)
    float*       __restrict__ out) {           // (B, N, L)
  __shared__ float a_lds[16 * LDS_A_STRIDE];
  __shared__ float x_lds[2][256 * LDS_X_STRIDE];

  const int b   = blockIdx.z;
  const int mt  = blockIdx.y;                  // 0..12 row tile
  const int n0  = blockIdx.x * 256;            // column base of this block
  const int tid = threadIdx.x;
  const int lane = tid & 31;
  const int wv   = tid >> 5;                   // wave 0..7
  const int hgrp = lane >> 4;                  // half-wave group 0/1
  const int lj   = lane & 15;                  // 0..15

  const float* xb    = inputs + (size_t)b * C_CH * N_NODES * LSEQ;  // channel 0
  const float* lap_b = lap    + (size_t)b * NPAD * NPAD;

  // ---- stage A strip: Lap rows [mt*16, mt*16+16), all K (padded with zeros) ----
  for (int i = tid; i < 16 * NPAD; i += 256) {
    const int m = i / NPAD;
    const int k = i - m * NPAD;
    a_lds[m * LDS_A_STRIDE + k] = lap_b[(size_t)(mt * 16 + m) * NPAD + k];
  }

  // ---- X chunk loader: 16 K-rows x 256 cols, stored transposed [n][k] ----
  auto load_chunk = [&](int buf, int kc) {
    const float* src = xb + (size_t)kc * LSEQ + n0 + tid;   // this thread's column
    float* dst = &x_lds[buf][tid * LDS_X_STRIDE];
#pragma unroll
    for (int kk = 0; kk < 16; ++kk) {
      const int k = kc + kk;
      dst[kk] = (k < N_NODES) ? src[(size_t)kk * LSEQ] : 0.0f;
    }
    if (kc + 16 < N_NODES)
      __builtin_prefetch((const void*)(src + (size_t)16 * LSEQ), 0, 0);  // global_prefetch_b8
  };

  v8f acc0 = {};
  v8f acc1 = {};

  load_chunk(0, 0);
  __syncthreads();

  int buf = 0;
  for (int kc = 0; kc < NPAD; kc += 16) {
    if (kc + 16 < NPAD) load_chunk(buf ^ 1, kc + 16);

#pragma unroll
    for (int ks = 0; ks < 4; ++ks) {
      const int kb = ks * 4;  // K offset within chunk
      // A fragment: lane l -> (m = lj, k = kb + 2*hgrp + {0,1})  (contiguous pair)
      const v2f af = *(const v2f*)&a_lds[lj * LDS_A_STRIDE + kc + kb + 2 * hgrp];
      // B fragments: lane l -> (n = lj, k = kb + 2*hgrp + {0,1}) in transposed LDS
      const v2f b0 = *(const v2f*)&x_lds[buf][(wv * 32 +      lj) * LDS_X_STRIDE + kb + 2 * hgrp];
      const v2f b1 = *(const v2f*)&x_lds[buf][(wv * 32 + 16 + lj) * LDS_X_STRIDE + kb + 2 * hgrp];
      acc0 = __builtin_amdgcn_wmma_f32_16x16x4_f32(false, af, false, b0, (short)0, acc0, false, false);
      acc1 = __builtin_amdgcn_wmma_f32_16x16x4_f32(false, af, false, b1, (short)0, acc1, false, false);
    }
    __syncthreads();
    buf ^= 1;
  }

  // ---- epilogue: out = acc + bias[m] + x[m][n] ----
  const int ii = ind[b];
  const int ntile0 = n0 + wv * 32;
#pragma unroll
  for (int r = 0; r < 8; ++r) {
    const int m = mt * 16 + r + 8 * hgrp;     // C/D layout: vgpr r, halfgroup h -> M = r + 8h
    if (m < N_NODES) {
      const float bias = bias_diffusion[(size_t)ii * N_NODES + m];
      const size_t row_in  = (size_t)m * LSEQ;
      const size_t row_out = (size_t)b * N_NODES * LSEQ + row_in;
      {
        const int n = ntile0 + lj;
        out[row_out + n] = acc0[r] + bias + xb[row_in + n];
      }
      {
        const int n = ntile0 + 16 + lj;
        out[row_out + n] = acc1[r] + bias + xb[row_in + n];
      }
    }
  }
}

// ---------------------------------------------------------------------------
extern "C" void kernel_launch(void* const* d_in, const int* in_sizes, int n_in,
                              void* d_out, int out_size, void* d_ws, size_t ws_size,
                              hipStream_t stream) {
  const float* inputs         = (const float*)d_in[0];
  const int*   ind            = (const int*)  d_in[1];
  const int*   rows           = (const int*)  d_in[2];
  const int*   cols           = (const int*)  d_in[3];
  const float* weight_diff    = (const float*)d_in[4];
  const float* bias_diffusion = (const float*)d_in[5];
  float*       out            = (float*)d_out;
  float*       lap            = (float*)d_ws;   // B * NPAD * NPAD * 4 = ~11.1 MB

  const int lap_elems = B_BATCH * NPAD * NPAD;
  zero_lap_kernel<<<(lap_elems + 255) / 256, 256, 0, stream>>>(lap, lap_elems);
  scatter_lap_kernel<<<B_BATCH, 256, 0, stream>>>(ind, rows, cols, weight_diff, lap);

  dim3 grid(LSEQ / 256, NPAD / 16, B_BATCH);  // (8, 13, 64)
  lap_gemm_kernel<<<grid, 256, 0, stream>>>(inputs, ind, bias_diffusion, lap, out);
}